// FiniteScalarQuantization_82480551952528
// MI455X (gfx1250) — compile-verified
//
#include <hip/hip_runtime.h>
#include <math.h>

#define NBLK 2048
#define NTHR 256
#define UNROLL 16       // 16 x b128 in flight per wave = 8 KB/wave of MLP
#define BOUND 512.0f    // NUM_LEVELS / 2

// ---- compile-time probes (stderr only, no codegen effect) ----
#if __has_builtin(__builtin_amdgcn_tanhf)
#warning "probe: __builtin_amdgcn_tanhf AVAILABLE (v_tanh_f32 path)"
#else
#warning "probe: __builtin_amdgcn_tanhf MISSING -> tanhf libm fallback"
#endif
// --------------------------------------------------------------

typedef float v4f __attribute__((ext_vector_type(4)));

__device__ __forceinline__ float hw_tanhf(float x) {
#if __has_builtin(__builtin_amdgcn_tanhf)
    return __builtin_amdgcn_tanhf(x);   // v_tanh_f32 (gfx1250 trans op)
#else
    return tanhf(x);
#endif
}

// Kernel 1: streaming quantize + per-block partial sum of (z - q)^2.
//   q = round(512 * tanh(z / T)) * T   (rint = nearest-even, matches jnp.round)
// Single-touch 268 MB stream vs 192 MB L2 -> NT temporal hints both directions.
// For this problem size each thread owns exactly UNROLL vec4s, so the main
// loop is one fully-batched pass: 16 independent b128 NT loads in flight.
__global__ void fsq_quant_kernel(const float* __restrict__ z,
                                 const float* __restrict__ temperature,
                                 float* __restrict__ out,
                                 float* __restrict__ partials,
                                 long long n)
{
    const float T    = temperature[0];
    const float invT = 1.0f / T;

    const v4f* __restrict__ z4 = reinterpret_cast<const v4f*>(z);
    v4f* __restrict__       o4 = reinterpret_cast<v4f*>(out);
    const long long n4 = n >> 2;
    const long long nT = (long long)gridDim.x * NTHR;   // total threads

    float acc = 0.0f;
    long long i = (long long)blockIdx.x * NTHR + threadIdx.x;

    for (; i + (UNROLL - 1) * nT < n4; i += UNROLL * nT) {
        v4f v[UNROLL];
#pragma unroll
        for (int j = 0; j < UNROLL; ++j) {
            v[j] = __builtin_nontemporal_load(&z4[i + j * nT]);   // b128 th:NT
        }
#pragma unroll
        for (int j = 0; j < UNROLL; ++j) {
            v4f q;
            q.x = rintf(BOUND * hw_tanhf(v[j].x * invT)) * T;
            q.y = rintf(BOUND * hw_tanhf(v[j].y * invT)) * T;
            q.z = rintf(BOUND * hw_tanhf(v[j].z * invT)) * T;
            q.w = rintf(BOUND * hw_tanhf(v[j].w * invT)) * T;
            float dx = v[j].x - q.x;
            float dy = v[j].y - q.y;
            float dz = v[j].z - q.z;
            float dw = v[j].w - q.w;
            acc = fmaf(dx, dx, acc);
            acc = fmaf(dy, dy, acc);
            acc = fmaf(dz, dz, acc);
            acc = fmaf(dw, dw, acc);
            __builtin_nontemporal_store(q, &o4[i + j * nT]);      // b128 th:NT
        }
    }

    // leftover vec4 iterations (generic-size safety)
    for (; i < n4; i += nT) {
        v4f v = __builtin_nontemporal_load(&z4[i]);
        v4f q;
        q.x = rintf(BOUND * hw_tanhf(v.x * invT)) * T;
        q.y = rintf(BOUND * hw_tanhf(v.y * invT)) * T;
        q.z = rintf(BOUND * hw_tanhf(v.z * invT)) * T;
        q.w = rintf(BOUND * hw_tanhf(v.w * invT)) * T;
        float dx = v.x - q.x;
        float dy = v.y - q.y;
        float dz = v.z - q.z;
        float dw = v.w - q.w;
        acc = fmaf(dx, dx, acc);
        acc = fmaf(dy, dy, acc);
        acc = fmaf(dz, dz, acc);
        acc = fmaf(dw, dw, acc);
        __builtin_nontemporal_store(q, &o4[i]);
    }

    // scalar tail (n not divisible by 4) — no-op here, kept for safety
    for (long long s = (n4 << 2) + (long long)blockIdx.x * NTHR + threadIdx.x; s < n; s += nT) {
        float v = z[s];
        float q = rintf(BOUND * hw_tanhf(v * invT)) * T;
        float d = v - q;
        acc = fmaf(d, d, acc);
        out[s] = q;
    }

    // deterministic block reduction (LDS tree)
    __shared__ float red[NTHR];
    red[threadIdx.x] = acc;
    __syncthreads();
    for (int s = NTHR / 2; s > 0; s >>= 1) {
        if ((int)threadIdx.x < s) red[threadIdx.x] += red[threadIdx.x + s];
        __syncthreads();
    }
    if (threadIdx.x == 0) partials[blockIdx.x] = red[0];
}

// Kernel 2: single block — fold partials, entropy term, emit scalar loss.
// total_loss = 2 * mean((z-q)^2) / T  - 0.1 * entropy_rate(usage_count)
__global__ void fsq_finalize_kernel(const float* __restrict__ partials, int nPartials,
                                    const float* __restrict__ usage, int K,
                                    const float* __restrict__ temperature,
                                    float* __restrict__ loss_out,
                                    double invN)
{
    __shared__ double sred[NTHR];
    const int t = threadIdx.x;

    // 1) sum of squared-error partials
    double s = 0.0;
    for (int i = t; i < nPartials; i += NTHR) s += (double)partials[i];
    sred[t] = s;
    __syncthreads();
    for (int k = NTHR / 2; k > 0; k >>= 1) {
        if (t < k) sred[t] += sred[t + k];
        __syncthreads();
    }
    const double sumSq = sred[0];
    __syncthreads();

    // 2) total usage
    double u = 0.0;
    for (int i = t; i < K; i += NTHR) u += (double)usage[i];
    sred[t] = u;
    __syncthreads();
    for (int k = NTHR / 2; k > 0; k >>= 1) {
        if (t < k) sred[t] += sred[t + k];
        __syncthreads();
    }
    const float total = (float)sred[0];
    __syncthreads();

    // 3) entropy: -sum(p * log2(p + 1e-10)), p = usage / max(total, 1e-10)
    const float denom = fmaxf(total, 1e-10f);
    double e = 0.0;
    for (int i = t; i < K; i += NTHR) {
        float p = usage[i] / denom;
        e += (double)(p * log2f(p + 1e-10f));
    }
    sred[t] = e;
    __syncthreads();
    for (int k = NTHR / 2; k > 0; k >>= 1) {
        if (t < k) sred[t] += sred[t + k];
        __syncthreads();
    }

    if (t == 0) {
        float ent = -(float)sred[0];
        if (!(total > 0.0f)) ent = 0.0f;
        const float entropy_loss = -ent * 0.1f;
        const float T = temperature[0];
        const float mean = (float)(sumSq * invN);
        loss_out[0] = (mean / T) * 2.0f + entropy_loss;
    }
}

extern "C" void kernel_launch(void* const* d_in, const int* in_sizes, int n_in,
                              void* d_out, int out_size, void* d_ws, size_t ws_size,
                              hipStream_t stream)
{
    const float* z           = (const float*)d_in[0];
    // d_in[1] (codebook) intentionally unused: the reference's cdist/argmin
    // ("assignments") never reaches either returned value — dead code.
    const float* temperature = (const float*)d_in[2];
    const float* usage       = (const float*)d_in[3];

    const long long n = (long long)in_sizes[0];   // 131072 * 256
    const int K       = in_sizes[3];              // 1024

    float* out  = (float*)d_out;
    float* loss = out + n;                        // out_size == n + 1
    float* partials = (float*)d_ws;               // NBLK floats of scratch

    fsq_quant_kernel<<<NBLK, NTHR, 0, stream>>>(z, temperature, out, partials, n);
    fsq_finalize_kernel<<<1, NTHR, 0, stream>>>(partials, NBLK, usage, K, temperature,
                                                loss, 1.0 / (double)n);
}